// Pointnet_2_9672266350688
// MI455X (gfx1250) — compile-verified
//
#include <hip/hip_runtime.h>
#include <math.h>
#include <stdint.h>

typedef float v2f __attribute__((ext_vector_type(2)));
typedef float v8f __attribute__((ext_vector_type(8)));

#define BATCH 8
#define EPS_BN 1e-5f

// ---------------------------------------------------------------------------
// Input normalization: per (batch, coord) mean/std over N points (std unbiased),
// xyz [B,3,N] -> l0 [B,N,3]
// ---------------------------------------------------------------------------
__global__ void normalize_kernel(const float* __restrict__ xyz, float* __restrict__ l0, int N) {
  int b = blockIdx.x / 3, c = blockIdx.x % 3;
  const float* src = xyz + ((size_t)b * 3 + c) * N;
  __shared__ float ssum[256], ssq[256];
  __shared__ float s_mean, s_inv;
  float s = 0.f, q = 0.f;
  for (int i = threadIdx.x; i < N; i += 256) { float v = src[i]; s += v; q += v * v; }
  ssum[threadIdx.x] = s; ssq[threadIdx.x] = q;
  __syncthreads();
  for (int st = 128; st > 0; st >>= 1) {
    if (threadIdx.x < st) { ssum[threadIdx.x] += ssum[threadIdx.x + st]; ssq[threadIdx.x] += ssq[threadIdx.x + st]; }
    __syncthreads();
  }
  if (threadIdx.x == 0) {
    float mean = ssum[0] / (float)N;
    float varu = (ssq[0] - (float)N * mean * mean) / (float)(N - 1);
    float sd = sqrtf(fmaxf(varu, 0.f));
    if (sd == 0.f) sd = 1e-8f;
    s_mean = mean; s_inv = 1.f / sd;
  }
  __syncthreads();
  float mean = s_mean, inv = s_inv;
  for (int i = threadIdx.x; i < N; i += 256)
    l0[((size_t)b * N + i) * 3 + c] = (src[i] - mean) * inv;
}

// ---------------------------------------------------------------------------
// Farthest point sampling: one block per batch; sequential over S iterations.
// ---------------------------------------------------------------------------
#define FPS_T 256
__global__ void fps_kernel(const float* __restrict__ xyz, int N, int S, int* __restrict__ idx_out) {
  int b = blockIdx.x;
  const float* P = xyz + (size_t)b * N * 3;
  __shared__ float dist[4096];
  __shared__ float smax[FPS_T];
  __shared__ int simax[FPS_T];
  __shared__ int s_far;
  int t = threadIdx.x;
  for (int i = t; i < N; i += FPS_T) dist[i] = 1e10f;
  if (t == 0) s_far = 0;
  __syncthreads();
  for (int it = 0; it < S; ++it) {
    int far = s_far;
    if (t == 0) idx_out[b * S + it] = far;
    float cx = P[far * 3 + 0], cy = P[far * 3 + 1], cz = P[far * 3 + 2];
    float bestv = -1.f; int besti = 0;
    for (int i = t; i < N; i += FPS_T) {
      float dx = P[i * 3] - cx, dy = P[i * 3 + 1] - cy, dz = P[i * 3 + 2] - cz;
      float d = dx * dx + dy * dy + dz * dz;
      float nd = fminf(dist[i], d);
      dist[i] = nd;
      if (nd > bestv) { bestv = nd; besti = i; }  // strict > keeps earliest i
    }
    smax[t] = bestv; simax[t] = besti;
    __syncthreads();
    for (int st = FPS_T / 2; st > 0; st >>= 1) {
      if (t < st) {
        float v2 = smax[t + st]; int i2 = simax[t + st];
        if (v2 > smax[t] || (v2 == smax[t] && i2 < simax[t])) { smax[t] = v2; simax[t] = i2; }
      }
      __syncthreads();
    }
    if (t == 0) s_far = simax[0];
    __syncthreads();
  }
}

// new_xyz[b,s,:] = xyz[b, idx[b,s], :]
__global__ void gather_kernel(const float* __restrict__ xyz, const int* __restrict__ idx,
                              int N, int S, float* __restrict__ nxyz) {
  int gid = blockIdx.x * blockDim.x + threadIdx.x;
  if (gid >= BATCH * S) return;
  int b = gid / S;
  int j = idx[gid];
  const float* p = xyz + ((size_t)b * N + j) * 3;
  nxyz[(size_t)gid * 3 + 0] = p[0];
  nxyz[(size_t)gid * 3 + 1] = p[1];
  nxyz[(size_t)gid * 3 + 2] = p[2];
}

// ---------------------------------------------------------------------------
// Ball query: first K indices (ascending) with d^2 <= r^2, backfill with first.
// ---------------------------------------------------------------------------
__global__ void ball_query_kernel(const float* __restrict__ xyz, const float* __restrict__ nxyz,
                                  int N, int S, int K, float r2, int* __restrict__ gidx) {
  int gid = blockIdx.x * blockDim.x + threadIdx.x;
  if (gid >= BATCH * S) return;
  int b = gid / S;
  const float* P = xyz + (size_t)b * N * 3;
  float qx = nxyz[(size_t)gid * 3 + 0];
  float qy = nxyz[(size_t)gid * 3 + 1];
  float qz = nxyz[(size_t)gid * 3 + 2];
  int* o = gidx + (size_t)gid * K;
  int cnt = 0, first = 0; bool found = false;
  for (int j = 0; j < N && cnt < K; ++j) {
    float dx = P[j * 3] - qx, dy = P[j * 3 + 1] - qy, dz = P[j * 3 + 2] - qz;
    if (dx * dx + dy * dy + dz * dz <= r2) {
      if (!found) { found = true; first = j; }
      o[cnt++] = j;
    }
  }
  for (; cnt < K; ++cnt) o[cnt] = first;
}

// Build grouped feature rows X[(b,s,k), 0:D+3] = [points[b,j,:], xyz[b,j,:]-new_xyz[b,s,:]]
__global__ void group_kernel(const float* __restrict__ xyz, const float* __restrict__ nxyz,
                             const float* __restrict__ points, const int* __restrict__ gidx,
                             int N, int S, int K, int D, float* __restrict__ X) {
  size_t gid = (size_t)blockIdx.x * blockDim.x + threadIdx.x;
  size_t total = (size_t)BATCH * S * K;
  if (gid >= total) return;
  size_t bs = gid / K;
  int b = (int)(bs / S);
  int j = gidx[gid];
  float* xr = X + gid * (D + 3);
  const float* pr = points + ((size_t)b * N + j) * D;
  for (int c = 0; c < D; ++c) xr[c] = pr[c];
  const float* pj = xyz + ((size_t)b * N + j) * 3;
  const float* q = nxyz + bs * 3;
  xr[D + 0] = pj[0] - q[0];
  xr[D + 1] = pj[1] - q[1];
  xr[D + 2] = pj[2] - q[2];
}

// ---------------------------------------------------------------------------
// GEMM via V_WMMA_F32_16X16X4_F32: Y[R,Cout] = X[R,Cin] @ W[Cout,Cin]^T
// One wave = one 16x16 output tile. R must be a multiple of 16 (true here).
//
// Branch-free inner loop:
//  - main loop covers Cin & ~3 with unconditional loads;
//  - B-side column overflow (o >= Cout) is handled by clamping the row
//    pointer only: WMMA keeps column n's products within lanes lane%16==n,
//    so garbage columns never leak into stored outputs;
//  - K tail zeroes only the A values (value select after a clamped load),
//    so garbage-free: 0 * finite = 0.
// ---------------------------------------------------------------------------
__global__ __launch_bounds__(128) void gemm_wmma_kernel(const float* __restrict__ X,
                                                        const float* __restrict__ W,
                                                        float* __restrict__ Y,
                                                        int R, int Cin, int Cout) {
  int wave = threadIdx.x >> 5;
  int lane = threadIdx.x & 31;
  int rowTile = blockIdx.x * 4 + wave;
  if (rowTile * 16 >= R) return;  // wave-uniform exit
  int colTile = blockIdx.y;
  int m = lane & 15;
  int half = lane >> 4;
  int row = rowTile * 16 + m;
  int o = colTile * 16 + m;
  int oc = (o < Cout) ? o : (Cout - 1);          // clamped: address always valid
  const float* xrow = X + (size_t)row * Cin;
  const float* wrow = W + (size_t)oc * Cin;
  v8f c = {0.f, 0.f, 0.f, 0.f, 0.f, 0.f, 0.f, 0.f};
  int kmain = Cin & ~3;
#pragma unroll 4
  for (int k0 = 0; k0 < kmain; k0 += 4) {
    int k = k0 + 2 * half;
    v2f a, b;
    a.x = xrow[k];
    a.y = xrow[k + 1];
    b.x = wrow[k];
    b.y = wrow[k + 1];
    c = __builtin_amdgcn_wmma_f32_16x16x4_f32(false, a, false, b, (short)0, c, false, false);
  }
  if (kmain < Cin) {
    int k = kmain + 2 * half;
    int k0c = (k < Cin) ? k : (Cin - 1);
    int k1c = (k + 1 < Cin) ? (k + 1) : (Cin - 1);
    float ax = xrow[k0c];
    float ay = xrow[k1c];
    ax = (k < Cin) ? ax : 0.f;          // value select (v_cndmask), load stays unconditional
    ay = (k + 1 < Cin) ? ay : 0.f;
    v2f a, b;
    a.x = ax; a.y = ay;
    b.x = wrow[k0c];                     // finite garbage OK: multiplied by 0
    b.y = wrow[k1c];
    c = __builtin_amdgcn_wmma_f32_16x16x4_f32(false, a, false, b, (short)0, c, false, false);
  }
  if (o < Cout) {
#pragma unroll
    for (int r = 0; r < 8; ++r) {
      int mm = r + 8 * half;
      Y[(size_t)(rowTile * 16 + mm) * Cout + o] = c[r];
    }
  }
}

// Per-channel mean / biased var over R rows. One block per channel,
// fixed-order tree reduction => deterministic.
__global__ void stats_kernel(const float* __restrict__ Y, int R, int C,
                             float* __restrict__ mean, float* __restrict__ var) {
  int c = blockIdx.x;
  __shared__ float ssum[256], ssq[256];
  float s = 0.f, q = 0.f;
  for (int r = threadIdx.x; r < R; r += 256) {
    float v = Y[(size_t)r * C + c];
    s += v; q += v * v;
  }
  ssum[threadIdx.x] = s; ssq[threadIdx.x] = q;
  __syncthreads();
  for (int st = 128; st > 0; st >>= 1) {
    if (threadIdx.x < st) { ssum[threadIdx.x] += ssum[threadIdx.x + st]; ssq[threadIdx.x] += ssq[threadIdx.x + st]; }
    __syncthreads();
  }
  if (threadIdx.x == 0) {
    float m = ssum[0] / (float)R;
    mean[c] = m;
    var[c] = fmaxf(ssq[0] / (float)R - m * m, 0.f);
  }
}

__global__ void bn_relu_kernel(float* __restrict__ Y, size_t total, int C,
                               const float* __restrict__ mean, const float* __restrict__ var,
                               const float* __restrict__ g, const float* __restrict__ bt) {
  size_t gid = (size_t)blockIdx.x * blockDim.x + threadIdx.x;
  if (gid >= total) return;
  int c = (int)(gid % C);
  float v = (Y[gid] - mean[c]) * rsqrtf(var[c] + EPS_BN) * g[c] + bt[c];
  Y[gid] = fmaxf(v, 0.f);
}

// Max over K group members: Y[(p,k),c] -> out[p, off+c] (out row stride Ctot)
__global__ void maxpool_kernel(const float* __restrict__ Y, int P, int K, int C,
                               float* __restrict__ out, int Ctot, int off) {
  size_t gid = (size_t)blockIdx.x * blockDim.x + threadIdx.x;
  if (gid >= (size_t)P * C) return;
  int c = (int)(gid % C);
  size_t p = gid / C;
  const float* yr = Y + p * (size_t)K * C + c;
  float m = yr[0];
  for (int k = 1; k < K; ++k) m = fmaxf(m, yr[(size_t)k * C]);
  out[p * Ctot + off + c] = m;
}

// ---------------------------------------------------------------------------
// Feature propagation: 3-NN inverse (squared) distance interpolation.
// ---------------------------------------------------------------------------
__global__ void fp_interp_kernel(const float* __restrict__ xyz1, const float* __restrict__ xyz2,
                                 const float* __restrict__ pts2, const float* __restrict__ pts1,
                                 int N1, int S2, int C1, int C2, float* __restrict__ X) {
  int gid = blockIdx.x * blockDim.x + threadIdx.x;
  if (gid >= BATCH * N1) return;
  int b = gid / N1;
  const float* q = xyz1 + (size_t)gid * 3;
  float qx = q[0], qy = q[1], qz = q[2];
  const float* P = xyz2 + (size_t)b * S2 * 3;
  float d0 = 3.4e38f, d1 = 3.4e38f, d2 = 3.4e38f;
  int i0 = 0, i1 = 0, i2 = 0;
  for (int j = 0; j < S2; ++j) {
    float dx = P[j * 3] - qx, dy = P[j * 3 + 1] - qy, dz = P[j * 3 + 2] - qz;
    float d = dx * dx + dy * dy + dz * dz;
    if (d < d0)      { d2 = d1; i2 = i1; d1 = d0; i1 = i0; d0 = d; i0 = j; }
    else if (d < d1) { d2 = d1; i2 = i1; d1 = d;  i1 = j; }
    else if (d < d2) { d2 = d;  i2 = j; }
  }
  float w0 = 1.f / (d0 + 1e-8f), w1 = 1.f / (d1 + 1e-8f), w2 = 1.f / (d2 + 1e-8f);
  float wsum = w0 + w1 + w2;
  w0 /= wsum; w1 /= wsum; w2 /= wsum;
  float* xr = X + (size_t)gid * (C1 + C2);
  if (pts1) {
    const float* pr = pts1 + (size_t)gid * C1;
    for (int c = 0; c < C1; ++c) xr[c] = pr[c];
  }
  const float* p0 = pts2 + ((size_t)b * S2 + i0) * C2;
  const float* p1 = pts2 + ((size_t)b * S2 + i1) * C2;
  const float* p2 = pts2 + ((size_t)b * S2 + i2) * C2;
  for (int c = 0; c < C2; ++c) xr[C1 + c] = w0 * p0[c] + w1 * p1[c] + w2 * p2[c];
}

// Head: logits = h @ W2^T + b2 ; log_softmax over 13 classes
__global__ void head2_kernel(const float* __restrict__ h, const float* __restrict__ W,
                             const float* __restrict__ bias, float* __restrict__ out, int total) {
  int gid = blockIdx.x * blockDim.x + threadIdx.x;
  if (gid >= total) return;
  const float* hr = h + (size_t)gid * 128;
  float acc[13];
  for (int o = 0; o < 13; ++o) {
    float s = bias[o];
    const float* wr = W + o * 128;
    for (int c = 0; c < 128; ++c) s += hr[c] * wr[c];
    acc[o] = s;
  }
  float mx = acc[0];
  for (int o = 1; o < 13; ++o) mx = fmaxf(mx, acc[o]);
  float se = 0.f;
  for (int o = 0; o < 13; ++o) se += expf(acc[o] - mx);
  float lse = logf(se) + mx;
  for (int o = 0; o < 13; ++o) out[(size_t)gid * 13 + o] = acc[o] - lse;
}

// ---------------------------------------------------------------------------
// Host orchestration
// ---------------------------------------------------------------------------
struct Lay { const float* W; const float* b; const float* bt; const float* g; int cin; int cout; };

static const int LCIN[35]  = {128,128,128, 352,256, 512,256, 1536,256, 128, 128,
                              6,16,16, 6,32,32, 99,64,64, 99,64,96,
                              259,128,196, 259,128,196, 515,256,256, 515,256,384};
static const int LCOUT[35] = {128,128,128, 256,128, 256,256, 256,256, 128, 13,
                              16,16,32, 32,32,64, 64,64,128, 64,96,128,
                              128,196,256, 128,196,256, 256,256,512, 256,384,512};

extern "C" void kernel_launch(void* const* d_in, const int* in_sizes, int n_in,
                              void* d_out, int out_size, void* d_ws, size_t ws_size,
                              hipStream_t stream) {
  (void)in_sizes; (void)out_size; (void)ws_size;
  if (n_in < 142) return;  // expect xyz, colors, 140 param leaves

  const float* xyz_in = (const float*)d_in[0];
  // colors (d_in[1]) unused by the reference forward

  Lay lays[35];
  for (int i = 0; i < 35; ++i) {
    lays[i].W  = (const float*)d_in[2 + i * 4 + 0];
    lays[i].b  = (const float*)d_in[2 + i * 4 + 1];
    lays[i].bt = (const float*)d_in[2 + i * 4 + 2];
    lays[i].g  = (const float*)d_in[2 + i * 4 + 3];
    lays[i].cin = LCIN[i]; lays[i].cout = LCOUT[i];
  }

  // --- workspace carve (deterministic bump allocator) ---
  uintptr_t base = (uintptr_t)d_ws;
  size_t off = 0;
  auto allocf = [&](size_t n) -> float* {
    float* p = (float*)(base + off);
    off = (off + n * sizeof(float) + 255) & ~(size_t)255;
    return p;
  };
  auto alloci = [&](size_t n) -> int* {
    int* p = (int*)(base + off);
    off = (off + n * sizeof(int) + 255) & ~(size_t)255;
    return p;
  };

  float* l0  = allocf((size_t)BATCH * 4096 * 3);
  float* l1x = allocf((size_t)BATCH * 1024 * 3);
  float* l1p = allocf((size_t)BATCH * 1024 * 96);
  float* l2x = allocf((size_t)BATCH * 256 * 3);
  float* l2p = allocf((size_t)BATCH * 256 * 256);
  float* l3x = allocf((size_t)BATCH * 64 * 3);
  float* l3p = allocf((size_t)BATCH * 64 * 512);
  float* l4x = allocf((size_t)BATCH * 16 * 3);
  float* l4p = allocf((size_t)BATCH * 16 * 1024);
  float* f4o = allocf((size_t)BATCH * 64 * 256);
  float* f3o = allocf((size_t)BATCH * 256 * 256);
  float* f2o = allocf((size_t)BATCH * 1024 * 128);
  float* mean = allocf(512);
  float* var  = allocf(512);
  float* bufA = allocf((size_t)262144 * 64);   // largest activation: 262144 x 64
  float* bufB = allocf((size_t)262144 * 64);
  int* fpsIdx = alloci((size_t)BATCH * 1024);
  int* gidx   = alloci((size_t)BATCH * 1024 * 32);

  // --- layer runner: GEMM (WMMA) -> deterministic BN stats -> BN+ReLU ---
  auto run_layer = [&](const float* X, float* Y, int R, int li) {
    const Lay& l = lays[li];
    dim3 grid((R / 16 + 3) / 4, (l.cout + 15) / 16);
    gemm_wmma_kernel<<<grid, 128, 0, stream>>>(X, l.W, Y, R, l.cin, l.cout);
    stats_kernel<<<l.cout, 256, 0, stream>>>(Y, R, l.cout, mean, var);
    size_t tot = (size_t)R * l.cout;
    bn_relu_kernel<<<(unsigned)((tot + 255) / 256), 256, 0, stream>>>(Y, tot, l.cout, mean, var, l.g, l.bt);
  };
  auto run_chain = [&](const float* X0, int R, int lbase, int nl) -> float* {
    const float* cur = X0;
    float* dst = (X0 == (const float*)bufA) ? bufB : bufA;
    for (int i = 0; i < nl; ++i) {
      run_layer(cur, dst, R, lbase + i);
      cur = dst;
      dst = (dst == bufA) ? bufB : bufA;
    }
    return (float*)cur;
  };

  // --- set abstraction stage (MSG, 2 branches) ---
  auto sa_stage = [&](const float* pxyz, const float* ppts, int N, int D, int S,
                      float rad1, int K1, int lb1, float rad2, int K2, int lb2,
                      float* oxyz, float* opts, int Ctot) {
    fps_kernel<<<BATCH, FPS_T, 0, stream>>>(pxyz, N, S, fpsIdx);
    int tot = BATCH * S;
    gather_kernel<<<(tot + 127) / 128, 128, 0, stream>>>(pxyz, fpsIdx, N, S, oxyz);
    float rads[2] = {rad1, rad2};
    int Ks[2] = {K1, K2}, lbs[2] = {lb1, lb2};
    int coff = 0;
    for (int br = 0; br < 2; ++br) {
      int K = Ks[br], lb = lbs[br];
      ball_query_kernel<<<(tot + 127) / 128, 128, 0, stream>>>(pxyz, oxyz, N, S, K,
                                                               rads[br] * rads[br], gidx);
      size_t rows = (size_t)tot * K;
      group_kernel<<<(unsigned)((rows + 127) / 128), 128, 0, stream>>>(pxyz, oxyz, ppts, gidx,
                                                                       N, S, K, D, bufA);
      float* res = run_chain(bufA, (int)rows, lb, 3);
      int cout = lays[lb + 2].cout;
      size_t oc = (size_t)tot * cout;
      maxpool_kernel<<<(unsigned)((oc + 127) / 128), 128, 0, stream>>>(res, tot, K, cout,
                                                                       opts, Ctot, coff);
      coff += cout;
    }
  };

  // --- feature propagation stage ---
  auto fp_stage = [&](const float* xyz1, const float* xyz2, const float* p1, int C1,
                      const float* p2, int C2, int N1, int S2, int lbase, int nl, float* outp) {
    int tot = BATCH * N1;
    fp_interp_kernel<<<(tot + 63) / 64, 64, 0, stream>>>(xyz1, xyz2, p2, p1, N1, S2, C1, C2, bufA);
    float* res = run_chain(bufA, tot, lbase, nl);
    int cout = lays[lbase + nl - 1].cout;
    hipMemcpyAsync(outp, res, (size_t)tot * cout * sizeof(float),
                   hipMemcpyDeviceToDevice, stream);
  };

  // ---------------- pipeline ----------------
  normalize_kernel<<<BATCH * 3, 256, 0, stream>>>(xyz_in, l0, 4096);

  sa_stage(l0,  l0,  4096,   3, 1024, 0.05f, 16, 11, 0.1f, 32, 14, l1x, l1p,   96);
  sa_stage(l1x, l1p, 1024,  96,  256, 0.10f, 16, 17, 0.2f, 32, 20, l2x, l2p,  256);
  sa_stage(l2x, l2p,  256, 256,   64, 0.20f, 16, 23, 0.4f, 32, 26, l3x, l3p,  512);
  sa_stage(l3x, l3p,   64, 512,   16, 0.40f, 16, 29, 0.8f, 32, 32, l4x, l4p, 1024);

  fp_stage(l3x, l4x, l3p, 512, l4p, 1024,   64,   16, 7, 2, f4o);
  fp_stage(l2x, l3x, l2p, 256, f4o,  256,  256,   64, 5, 2, f3o);
  fp_stage(l1x, l2x, l1p,  96, f3o,  256, 1024,  256, 3, 2, f2o);

  // fp1 (no points1) + head, kept in ping-pong buffers
  int tot0 = BATCH * 4096;
  fp_interp_kernel<<<(tot0 + 63) / 64, 64, 0, stream>>>(l0, l1x, f2o, nullptr,
                                                        4096, 1024, 0, 128, bufA);
  float* res = run_chain(bufA, tot0, 0, 3);            // ends in bufB
  float* hbuf = (res == bufA) ? bufB : bufA;
  run_layer(res, hbuf, tot0, 9);                       // head1 conv+bn+relu
  head2_kernel<<<(tot0 + 63) / 64, 64, 0, stream>>>(hbuf, lays[10].W, lays[10].b,
                                                    (float*)d_out, tot0);
}